// SimpleGATSingleHeadLayer_83476984365538
// MI455X (gfx1250) — compile-verified
//
#include <hip/hip_runtime.h>
#include <hip/hip_bf16.h>

typedef float v2f __attribute__((ext_vector_type(2)));
typedef float v8f __attribute__((ext_vector_type(8)));

#define DIN  128
#define DOUT 64

// ---------------------------------------------------------------------------
// Kernel 1: z = h @ W  via V_WMMA_F32_16X16X4_F32.
// Block = 128 threads (4 waves). Each wave computes a 16x64 tile of z.
// W is staged into LDS K-pair-interleaved:  Wp[(k>>1)*128 + n*2 + (k&1)] =
// W[k][n], so each B fragment {W[k][n], W[k+1][n]} is a single aligned
// ds_load_b64 into a consecutive VGPR pair (no repacking moves).
// Each wave stages its 16x128 slice of h into LDS with coalesced float4 loads;
// the A fragment {h[m][k], h[m][k+1]} is contiguous row-major -> one b64 load.
// ---------------------------------------------------------------------------
__global__ void __launch_bounds__(128)
gat_wmma_gemm_kernel(const float* __restrict__ h, const float* __restrict__ W,
                     float* __restrict__ z, int N) {
    __shared__ float Wl[DIN * DOUT];      // 32 KB, pair-interleaved layout
    __shared__ float Al[4][16 * DIN];     // 4 waves * 8 KB = 32 KB

    const int tid  = threadIdx.x;
    const int lane = tid & 31;
    const int wave = tid >> 5;
    const int row0 = blockIdx.x * 64 + wave * 16;

    // Stage W into LDS (pair-interleaved): 2048 float4 / 128 threads.
    {
        const float4* Wv = (const float4*)W;
        #pragma unroll
        for (int it = 0; it < 16; ++it) {
            int idx  = tid + it * 128;        // float4 index
            int flat = idx * 4;
            int k = flat >> 6;                // row of W (0..127)
            int n = flat & 63;                // col of W (multiple of 4)
            float4 v = Wv[idx];
            float* dst = Wl + (k >> 1) * 128 + (k & 1);
            dst[(n + 0) * 2] = v.x;
            dst[(n + 1) * 2] = v.y;
            dst[(n + 2) * 2] = v.z;
            dst[(n + 3) * 2] = v.w;
        }
    }
    // Stage this wave's 16x128 A tile (clamp rows for the tail block).
    {
        float4* Av = (float4*)Al[wave];
        #pragma unroll
        for (int i = lane; i < 512; i += 32) {
            int flat = i * 4;
            int r = flat >> 7;          // 0..15
            int c = flat & 127;         // multiple of 4 -> 16B aligned
            int rr = row0 + r;
            if (rr >= N) rr = N - 1;    // clamped, stores are guarded below
            Av[i] = *(const float4*)(h + (size_t)rr * DIN + c);
        }
    }
    __syncthreads();

    // Accumulators: four 16x16 f32 tiles covering N = 0..63.
    v8f c0 = {}, c1 = {}, c2 = {}, c3 = {};

    const int m  = lane & 15;            // row/col within tile (both halves)
    const int kb = (lane >> 4) << 1;     // lanes 16..31 carry K+2, K+3
    const float* Aw = Al[wave] + m * DIN + kb;

    #pragma unroll
    for (int kk = 0; kk < DIN; kk += 4) {
        // A fragment: 16x4, contiguous {h[m][k], h[m][k+1]} -> one b64 load.
        v2f af = *(const v2f*)(Aw + kk);
        // B fragments: pair-interleaved rows -> one b64 load each.
        const float* Wp = Wl + ((kk + kb) >> 1) * 128 + m * 2;
        v2f b0 = *(const v2f*)(Wp + 0);    // N tile 0 (cols  0..15)
        v2f b1 = *(const v2f*)(Wp + 32);   // N tile 1 (cols 16..31)
        v2f b2 = *(const v2f*)(Wp + 64);   // N tile 2 (cols 32..47)
        v2f b3 = *(const v2f*)(Wp + 96);   // N tile 3 (cols 48..63)
        c0 = __builtin_amdgcn_wmma_f32_16x16x4_f32(false, af, false, b0, (short)0, c0, false, false);
        c1 = __builtin_amdgcn_wmma_f32_16x16x4_f32(false, af, false, b1, (short)0, c1, false, false);
        c2 = __builtin_amdgcn_wmma_f32_16x16x4_f32(false, af, false, b2, (short)0, c2, false, false);
        c3 = __builtin_amdgcn_wmma_f32_16x16x4_f32(false, af, false, b3, (short)0, c3, false, false);
    }

    // Store D: lanes 0-15 hold rows M=v, lanes 16-31 rows M=v+8; N = tile*16 + (lane&15).
    const int half = (lane >> 4) << 3;
    #pragma unroll
    for (int v = 0; v < 8; ++v) {
        int grow = row0 + v + half;
        if (grow < N) {
            float* zp = z + (size_t)grow * DOUT + m;
            zp[0]  = c0[v];
            zp[16] = c1[v];
            zp[32] = c2[v];
            zp[48] = c3[v];
        }
    }
}

// ---------------------------------------------------------------------------
// Kernel 2: per-node attention scalars  es[n] = z[n]·a[:64], ed[n] = z[n]·a[64:]
// One wave32 per node: each lane takes a float2 (coalesced 256B per wave).
// ---------------------------------------------------------------------------
__global__ void __launch_bounds__(256)
gat_node_dots_kernel(const float* __restrict__ z, const float* __restrict__ av,
                     float* __restrict__ es, float* __restrict__ ed, int N) {
    int t = blockIdx.x * blockDim.x + threadIdx.x;
    int node = t >> 5;
    int lane = t & 31;
    if (node >= N) return;
    float2 zz = ((const float2*)(z + (size_t)node * DOUT))[lane];
    float d0 = zz.x * av[2 * lane]      + zz.y * av[2 * lane + 1];
    float d1 = zz.x * av[DOUT + 2*lane] + zz.y * av[DOUT + 2*lane + 1];
    #pragma unroll
    for (int o = 16; o > 0; o >>= 1) {
        d0 += __shfl_down(d0, o, 32);
        d1 += __shfl_down(d1, o, 32);
    }
    if (lane == 0) { es[node] = d0; ed[node] = d1; }
}

// Monotone order-preserving float<->uint mapping for atomic f32 max.
__device__ __forceinline__ unsigned ford(float x) {
    unsigned u = __float_as_uint(x);
    return (u & 0x80000000u) ? ~u : (u | 0x80000000u);
}
__device__ __forceinline__ float funo(unsigned u) {
    unsigned b = (u & 0x80000000u) ? (u ^ 0x80000000u) : ~u;
    return __uint_as_float(b);
}

// ---------------------------------------------------------------------------
// Kernel 3: e[i] = es[src[i]] + ed[dst[i]];  segment-max via atomic u32 max.
// ---------------------------------------------------------------------------
__global__ void __launch_bounds__(256)
gat_edge_logit_kernel(const int* __restrict__ src, const int* __restrict__ dst,
                      const float* __restrict__ es, const float* __restrict__ ed,
                      float* __restrict__ e, unsigned* __restrict__ mu, int E) {
    int i = blockIdx.x * blockDim.x + threadIdx.x;
    if (i >= E) return;
    int d = dst[i];
    float v = es[src[i]] + ed[d];
    e[i] = v;
    atomicMax(&mu[d], ford(v));
}

// ---------------------------------------------------------------------------
// Kernel 4: w[i] = exp(e[i] - m[dst]);  s[dst] += w  (w overwrites e in place).
// ---------------------------------------------------------------------------
__global__ void __launch_bounds__(256)
gat_edge_exp_kernel(const int* __restrict__ dst, float* __restrict__ e,
                    const unsigned* __restrict__ mu, float* __restrict__ s, int E) {
    int i = blockIdx.x * blockDim.x + threadIdx.x;
    if (i >= E) return;
    int d = dst[i];
    float w = __expf(e[i] - funo(mu[d]));
    e[i] = w;
    atomicAdd(&s[d], w);
}

// ---------------------------------------------------------------------------
// Kernel 5: out[dst] += (w/s[dst]) * z[src].  64 threads per edge, one f32
// atomic per (edge, col). z and out are L2-resident (25.6 MB each << 192 MB).
// ---------------------------------------------------------------------------
__global__ void __launch_bounds__(256)
gat_scatter_kernel(const int* __restrict__ src, const int* __restrict__ dst,
                   const float* __restrict__ w, const float* __restrict__ s,
                   const float* __restrict__ z, float* __restrict__ out, int E) {
    int t = blockIdx.x * blockDim.x + threadIdx.x;
    int i = t >> 6;
    if (i >= E) return;
    int col = t & 63;
    int sn = src[i], dn = dst[i];
    float alpha = w[i] / s[dn];
    atomicAdd(&out[(size_t)dn * DOUT + col], alpha * z[(size_t)sn * DOUT + col]);
}

static inline size_t align256(size_t x) { return (x + 255) & ~(size_t)255; }

extern "C" void kernel_launch(void* const* d_in, const int* in_sizes, int n_in,
                              void* d_out, int out_size, void* d_ws, size_t ws_size,
                              hipStream_t stream) {
    const float* h  = (const float*)d_in[0];
    const float* W  = (const float*)d_in[1];
    const float* av = (const float*)d_in[2];
    const int* src  = (const int*)d_in[3];
    const int* dst  = (const int*)d_in[4];
    float* out = (float*)d_out;

    const int N = in_sizes[0] / DIN;
    const int E = in_sizes[3];

    // Workspace carve-out.
    char* p = (char*)d_ws;
    float*    z  = (float*)p;    p += align256((size_t)N * DOUT * sizeof(float));
    float*    es = (float*)p;    p += align256((size_t)N * sizeof(float));
    float*    ed = (float*)p;    p += align256((size_t)N * sizeof(float));
    unsigned* mu = (unsigned*)p; p += align256((size_t)N * sizeof(unsigned));
    float*    s  = (float*)p;    p += align256((size_t)N * sizeof(float));
    float*    e  = (float*)p;    p += align256((size_t)E * sizeof(float));

    // Zero-init accumulation targets (graph-capturable async memsets).
    hipMemsetAsync(out, 0, (size_t)N * DOUT * sizeof(float), stream);
    hipMemsetAsync(mu,  0, (size_t)N * sizeof(unsigned), stream);  // mapped -inf
    hipMemsetAsync(s,   0, (size_t)N * sizeof(float), stream);

    // 1) projection via WMMA
    gat_wmma_gemm_kernel<<<(N + 63) / 64, 128, 0, stream>>>(h, W, z, N);
    // 2) per-node attention scalars
    {
        long long threads = (long long)N * 32;
        gat_node_dots_kernel<<<(int)((threads + 255) / 256), 256, 0, stream>>>(z, av, es, ed, N);
    }
    // 3) edge logits + segment max
    gat_edge_logit_kernel<<<(E + 255) / 256, 256, 0, stream>>>(src, dst, es, ed, e, mu, E);
    // 4) exp + segment sum
    gat_edge_exp_kernel<<<(E + 255) / 256, 256, 0, stream>>>(dst, e, mu, s, E);
    // 5) weighted scatter
    {
        long long threads = (long long)E * 64;
        gat_scatter_kernel<<<(int)((threads + 255) / 256), 256, 0, stream>>>(src, dst, e, s, z, out, E);
    }
}